// TrainableFIS_26465588478388
// MI455X (gfx1250) — compile-verified
//
#include <hip/hip_runtime.h>

typedef __attribute__((ext_vector_type(2))) float v2f;
typedef __attribute__((ext_vector_type(8))) float v8f;

__device__ __forceinline__ float sel3(int m, float a, float b, float c) {
    // branch-free 3-way select (lowers to v_cndmask pairs)
    return m == 0 ? a : (m == 1 ? b : c);
}

__global__ __launch_bounds__(256) void TrainableFIS_kernel(
    const float* __restrict__ x,          // (B,4)
    const float* __restrict__ in_params,  // (4,3,2) [mean,sigma]
    const float* __restrict__ out_params, // (3,2)   [mean,sigma]
    const int*   __restrict__ rules_ant,  // (81,4)
    const int*   __restrict__ rules_con,  // (81,)
    float* __restrict__ y,                // (B,)
    int B)
{
    // padded output-MF table on the defuzz grid: 101 points -> 104 (pad = 0
    // so padded points contribute agg = max_k min(F_k, 0) = 0)
    __shared__ float sOMU[3][104];
    __shared__ int   sRule[82];           // 82nd entry = no-op rule (con=3)

    const int tid = threadIdx.x;

    if (tid < 81) {
        int a0 = rules_ant[tid * 4 + 0];
        int a1 = rules_ant[tid * 4 + 1];
        int a2 = rules_ant[tid * 4 + 2];
        int a3 = rules_ant[tid * 4 + 3];
        int cn = rules_con[tid];
        sRule[tid] = a0 | (a1 << 2) | (a2 << 4) | (a3 << 6) | (cn << 8);
    }
    if (tid == 81) sRule[81] = (3 << 8);  // con=3 matches no class -> no-op
    for (int idx = tid; idx < 3 * 104; idx += 256) {
        int k = idx / 104, p = idx % 104;
        float v = 0.0f;
        if (p < 101) {
            float g = (float)p / 100.0f;
            float z = (g - out_params[k * 2 + 0]) / out_params[k * 2 + 1];
            v = __expf(-0.5f * z * z);
        }
        sOMU[k][p] = v;
    }
    __syncthreads();

    const int lane = tid & 31;
    const int wave = tid >> 5;
    const int row  = lane & 15;   // tile row this lane owns
    const int half = lane >> 4;   // 0: K=0,1  1: K=2,3  (A-matrix lane halves)
    const int tileBase = blockIdx.x * 128 + wave * 16;
    int b  = tileBase + row;
    int bc = (b < B) ? b : (B - 1);

    // ---- input membership degrees mu[d][m] (12 exps) ----
    float4 xv4 = *(const float4*)(x + bc * 4);
    float xv[4] = {xv4.x, xv4.y, xv4.z, xv4.w};
    float mu[4][3];
    #pragma unroll
    for (int d = 0; d < 4; ++d) {
        #pragma unroll
        for (int m = 0; m < 3; ++m) {
            float mean = in_params[(d * 3 + m) * 2 + 0];
            float sig  = in_params[(d * 3 + m) * 2 + 1];
            float z = (xv[d] - mean) / sig;
            mu[d][m] = __expf(-0.5f * z * z);
        }
    }

    // ---- group-max firing: F[k] = max_{r: con==k} min_d mu[d][ant_rd] ----
    // max-min distributivity collapses the B*R*P composition to B*K*P.
    // Lane L and L+16 own the same row, so split the 82 (padded) rules
    // across the wave halves (41 each) and merge with one shfl_xor(16).
    float F0 = 0.0f, F1 = 0.0f, F2 = 0.0f;
    const int rbase = half * 41;
    for (int r = 0; r < 41; ++r) {
        int rr = sRule[rbase + r];
        float f =    sel3( rr       & 3, mu[0][0], mu[0][1], mu[0][2]);
        f = fminf(f, sel3((rr >> 2) & 3, mu[1][0], mu[1][1], mu[1][2]));
        f = fminf(f, sel3((rr >> 4) & 3, mu[2][0], mu[2][1], mu[2][2]));
        f = fminf(f, sel3((rr >> 6) & 3, mu[3][0], mu[3][1], mu[3][2]));
        int cn = (rr >> 8) & 3;
        F0 = (cn == 0) ? fmaxf(F0, f) : F0;
        F1 = (cn == 1) ? fmaxf(F1, f) : F1;
        F2 = (cn == 2) ? fmaxf(F2, f) : F2;
    }
    F0 = fmaxf(F0, __shfl_xor(F0, 16, 32));
    F1 = fmaxf(F1, __shfl_xor(F1, 16, 32));
    F2 = fmaxf(F2, __shfl_xor(F2, 16, 32));

    // ---- defuzzification via V_WMMA_F32_16X16X4_F32 ----
    // A[m][k] = agg[tile row m][point p0+k]   (16x4 f32, 2 VGPRs/lane)
    // B[k][0] = grid[p0+k], B[k][1] = 1.0     (4x16 f32, 2 VGPRs/lane)
    // C[:,0] accumulates numerator, C[:,1] accumulates denominator.
    const int   poff  = half * 2;                  // lane's K offset in chunk
    const int   ncol  = row;                       // lane's N column for B
    const float selg  = (ncol == 0) ? 1.0f : 0.0f; // column 0 carries grid
    const float sel1  = (ncol == 1) ? 1.0f : 0.0f; // column 1 carries ones
    const float gbase = (float)poff * 0.01f;       // grid step = 1/100
    v8f c = {0.f, 0.f, 0.f, 0.f, 0.f, 0.f, 0.f, 0.f};

    #pragma unroll
    for (int ch = 0; ch < 26; ++ch) {
        int pA = ch * 4 + poff;

        float a0 = fmaxf(fmaxf(fminf(F0, sOMU[0][pA]),
                               fminf(F1, sOMU[1][pA])),
                               fminf(F2, sOMU[2][pA]));
        float a1 = fmaxf(fmaxf(fminf(F0, sOMU[0][pA + 1]),
                               fminf(F1, sOMU[1][pA + 1])),
                               fminf(F2, sOMU[2][pA + 1]));
        v2f A; A.x = a0; A.y = a1;

        float g0 = gbase + (float)ch * 0.04f;      // constant under unroll
        float g1 = g0 + 0.01f;
        v2f Bv;
        Bv.x = fmaf(g0, selg, sel1);               // branchless column build
        Bv.y = fmaf(g1, selg, sel1);

        c = __builtin_amdgcn_wmma_f32_16x16x4_f32(
                /*neg_a=*/false, A, /*neg_b=*/false, Bv,
                /*c_mod=*/(short)0, c, /*reuse_a=*/false, /*reuse_b=*/false);
    }

    // C layout (16x16 f32): VGPR i, lanes 0-15 -> C[i][lane]; lanes 16-31 -> C[i+8][lane-16]
    // lane 0 : numerators rows 0..7    lane 16: numerators rows 8..15
    // lane 1 : denominators rows 0..7  lane 17: denominators rows 8..15
    float num[8], den[8];
    #pragma unroll
    for (int i = 0; i < 8; ++i) {
        num[i] = c[i];
        den[i] = __shfl_xor(c[i], 1, 32);
    }
    if ((lane & 15) == 0) {
        int base = tileBase + half * 8;
        #pragma unroll
        for (int i = 0; i < 8; ++i) {
            int ob = base + i;
            if (ob < B) y[ob] = num[i] / (den[i] + 1e-12f);
        }
    }
}

extern "C" void kernel_launch(void* const* d_in, const int* in_sizes, int n_in,
                              void* d_out, int out_size, void* d_ws, size_t ws_size,
                              hipStream_t stream) {
    const float* x          = (const float*)d_in[0];
    const float* in_params  = (const float*)d_in[1];
    const float* out_params = (const float*)d_in[2];
    const int*   rules_ant  = (const int*)d_in[3];
    const int*   rules_con  = (const int*)d_in[4];
    // d_in[5] = point_n (101), baked into the kernel's grid/table sizes
    float* y = (float*)d_out;

    int B = in_sizes[0] / 4;                // (B, D=4)
    int blocks = (B + 127) / 128;           // 8 waves x 16 rows per block
    TrainableFIS_kernel<<<blocks, 256, 0, stream>>>(
        x, in_params, out_params, rules_ant, rules_con, y, B);
}